// GNNEncoder_24283745091695
// MI455X (gfx1250) — compile-verified
//
#include <hip/hip_runtime.h>
#include <math.h>

typedef float v2f __attribute__((ext_vector_type(2)));
typedef float v8f __attribute__((ext_vector_type(8)));

#ifndef INFINITY
#define INFINITY __builtin_inff()
#endif

// ---------------------------------------------------------------------------
// fill kernel (grid-stride)
// ---------------------------------------------------------------------------
__global__ void k_fill(float* __restrict__ p, float v, long long n) {
    long long i = (long long)blockIdx.x * blockDim.x + threadIdx.x;
    long long stride = (long long)gridDim.x * blockDim.x;
    for (; i < n; i += stride) p[i] = v;
}

// ---------------------------------------------------------------------------
// degree accumulation: deg[dst[e]] += 1 (deg pre-initialized to 1.0 = self loop)
// ---------------------------------------------------------------------------
__global__ void k_deg(const int* __restrict__ dst, float* __restrict__ deg, int E) {
    int i = blockIdx.x * blockDim.x + threadIdx.x;
    if (i < E) atomicAdd(&deg[dst[i]], 1.0f);
}

__global__ void k_rsqrt(float* __restrict__ d, int N) {
    int i = blockIdx.x * blockDim.x + threadIdx.x;
    if (i < N) d[i] = rsqrtf(d[i]);
}

// ---------------------------------------------------------------------------
// GEMM: Out[N,64] = A[N,64] @ W[64,64]  using V_WMMA_F32_16X16X4_F32
// Block = 256 threads = 8 waves; each wave computes a 16x64 output tile.
// W is staged transposed in LDS so B fragments are contiguous 8B loads.
// A rows for out-of-range lanes are clamped to row 0 (loads always in
// bounds); their garbage D rows are simply never stored. This keeps the
// whole kernel branch-free until after the last WMMA (EXEC all-ones).
// ---------------------------------------------------------------------------
__global__ void k_gemm64(const float* __restrict__ A, const float* __restrict__ W,
                         float* __restrict__ Out, int N) {
    __shared__ float Wt[64 * 64];  // Wt[n*64 + k] = W[k*64 + n]
    int t = threadIdx.x;
#pragma unroll
    for (int i = 0; i < 16; ++i) {
        int idx = t + i * 256;
        int k = idx >> 6, n = idx & 63;
        Wt[n * 64 + k] = W[idx];
    }
    __syncthreads();

    int wave  = t >> 5;
    int lane  = t & 31;
    int m     = lane & 15;      // M index within tile
    int khalf = lane >> 4;      // selects K pair {0,1} vs {2,3}
    int tile  = blockIdx.x * 8 + wave;
    int row   = tile * 16 + m;
    int crow  = row < N ? row : 0;           // clamp: load is always in-bounds
    const float* arow = A + (long long)crow * 64;

    // Prefetch the 16 A fragments (16 x global_load_b64 in flight).
    v2f afrag[16];
#pragma unroll
    for (int kk = 0; kk < 16; ++kk)
        afrag[kk] = *(const v2f*)(arow + kk * 4 + khalf * 2);

    v8f acc0 = {}, acc1 = {}, acc2 = {}, acc3 = {};
    int n0 = lane & 15;

#pragma unroll
    for (int kk = 0; kk < 16; ++kk) {
        int kb = kk * 4 + khalf * 2;
        v2f b0, b1, b2, b3;
        b0.x = Wt[(n0 +  0) * 64 + kb]; b0.y = Wt[(n0 +  0) * 64 + kb + 1];
        b1.x = Wt[(n0 + 16) * 64 + kb]; b1.y = Wt[(n0 + 16) * 64 + kb + 1];
        b2.x = Wt[(n0 + 32) * 64 + kb]; b2.y = Wt[(n0 + 32) * 64 + kb + 1];
        b3.x = Wt[(n0 + 48) * 64 + kb]; b3.y = Wt[(n0 + 48) * 64 + kb + 1];
        acc0 = __builtin_amdgcn_wmma_f32_16x16x4_f32(false, afrag[kk], false, b0, (short)0, acc0, false, false);
        acc1 = __builtin_amdgcn_wmma_f32_16x16x4_f32(false, afrag[kk], false, b1, (short)0, acc1, false, false);
        acc2 = __builtin_amdgcn_wmma_f32_16x16x4_f32(false, afrag[kk], false, b2, (short)0, acc2, false, false);
        acc3 = __builtin_amdgcn_wmma_f32_16x16x4_f32(false, afrag[kk], false, b3, (short)0, acc3, false, false);
    }

    // D layout: VGPR r -> M = r + 8*khalf, N = lane&15 (+16*ntile)
    int col = lane & 15;
    int rowBase = tile * 16 + khalf * 8;
#pragma unroll
    for (int r = 0; r < 8; ++r) {
        int orow = rowBase + r;
        if (orow < N) {
            float* o = Out + (long long)orow * 64 + col;
            o[0]  = acc0[r];
            o[16] = acc1[r];
            o[32] = acc2[r];
            o[48] = acc3[r];
        }
    }
}

// ---------------------------------------------------------------------------
// edge aggregation: out[dst] += hW[src] * dinv[src]*dinv[dst]
// one wave (32 lanes) per edge; 2 columns per lane. Edges E..E+N-1 are self loops.
// ---------------------------------------------------------------------------
__global__ void k_aggregate(const float* __restrict__ hW, const int* __restrict__ src,
                            const int* __restrict__ dst, const float* __restrict__ dinv,
                            float* __restrict__ out, int E, int N) {
    int e = blockIdx.x * (blockDim.x >> 5) + (threadIdx.x >> 5);
    int lane = threadIdx.x & 31;
    int total = E + N;
    if (e >= total) return;
    int s, d;
    if (e < E) { s = src[e]; d = dst[e]; } else { s = e - E; d = s; }
    float norm = dinv[s] * dinv[d];
    const float* hs = hW + (long long)s * 64;
    float* od = out + (long long)d * 64;
    atomicAdd(&od[lane],      hs[lane]      * norm);
    atomicAdd(&od[lane + 32], hs[lane + 32] * norm);
}

// ---------------------------------------------------------------------------
// bias + relu, in place over [N,64]
// ---------------------------------------------------------------------------
__global__ void k_bias_relu(float* __restrict__ h, const float* __restrict__ b, long long total) {
    long long i = (long long)blockIdx.x * blockDim.x + threadIdx.x;
    long long stride = (long long)gridDim.x * blockDim.x;
    for (; i < total; i += stride) {
        float v = h[i] + b[i & 63];
        h[i] = v > 0.0f ? v : 0.0f;
    }
}

// ---------------------------------------------------------------------------
// scores[i] = dot(h[i,:], att_w) ; one wave per node, shuffle reduction
// ---------------------------------------------------------------------------
__global__ void k_scores(const float* __restrict__ h, const float* __restrict__ aw,
                         float* __restrict__ s, int N) {
    int node = blockIdx.x * (blockDim.x >> 5) + (threadIdx.x >> 5);
    int lane = threadIdx.x & 31;
    if (node >= N) return;
    const float* hr = h + (long long)node * 64;
    float v = hr[lane] * aw[lane] + hr[lane + 32] * aw[lane + 32];
#pragma unroll
    for (int off = 16; off > 0; off >>= 1) v += __shfl_xor(v, off, 32);
    if (lane == 0) s[node] = v;
}

// ---------------------------------------------------------------------------
// softmax reductions
// ---------------------------------------------------------------------------
__global__ void k_max_part(const float* __restrict__ s, float* __restrict__ part, int N) {
    __shared__ float sm[256];
    float m = -INFINITY;
    for (int i = blockIdx.x * blockDim.x + threadIdx.x; i < N; i += gridDim.x * blockDim.x)
        m = fmaxf(m, s[i]);
    sm[threadIdx.x] = m;
    __syncthreads();
    for (int o = 128; o > 0; o >>= 1) {
        if (threadIdx.x < o) sm[threadIdx.x] = fmaxf(sm[threadIdx.x], sm[threadIdx.x + o]);
        __syncthreads();
    }
    if (threadIdx.x == 0) part[blockIdx.x] = sm[0];
}

__global__ void k_max_final(const float* __restrict__ part, float* __restrict__ g, int n) {
    __shared__ float sm[256];
    float m = -INFINITY;
    for (int i = threadIdx.x; i < n; i += 256) m = fmaxf(m, part[i]);
    sm[threadIdx.x] = m;
    __syncthreads();
    for (int o = 128; o > 0; o >>= 1) {
        if (threadIdx.x < o) sm[threadIdx.x] = fmaxf(sm[threadIdx.x], sm[threadIdx.x + o]);
        __syncthreads();
    }
    if (threadIdx.x == 0) g[0] = sm[0];
}

__global__ void k_exp_part(float* __restrict__ s, const float* __restrict__ g,
                           float* __restrict__ part, int N) {
    __shared__ float sm[256];
    float gmax = g[0];
    float acc = 0.0f;
    for (int i = blockIdx.x * blockDim.x + threadIdx.x; i < N; i += gridDim.x * blockDim.x) {
        float e = expf(s[i] - gmax);
        s[i] = e;
        acc += e;
    }
    sm[threadIdx.x] = acc;
    __syncthreads();
    for (int o = 128; o > 0; o >>= 1) {
        if (threadIdx.x < o) sm[threadIdx.x] += sm[threadIdx.x + o];
        __syncthreads();
    }
    if (threadIdx.x == 0) part[blockIdx.x] = sm[0];
}

__global__ void k_sum_final(const float* __restrict__ part, float* __restrict__ g, int n) {
    __shared__ float sm[256];
    float acc = 0.0f;
    for (int i = threadIdx.x; i < n; i += 256) acc += part[i];
    sm[threadIdx.x] = acc;
    __syncthreads();
    for (int o = 128; o > 0; o >>= 1) {
        if (threadIdx.x < o) sm[threadIdx.x] += sm[threadIdx.x + o];
        __syncthreads();
    }
    if (threadIdx.x == 0) g[1] = sm[0];
}

// attended = h * e[node] / gsum  (in place over h)
__global__ void k_attend(float* __restrict__ h, const float* __restrict__ e,
                         const float* __restrict__ g, long long total) {
    float inv = 1.0f / g[1];
    long long i = (long long)blockIdx.x * blockDim.x + threadIdx.x;
    long long stride = (long long)gridDim.x * blockDim.x;
    for (; i < total; i += stride) h[i] = h[i] * e[i >> 6] * inv;
}

// probs = e / gsum (in place)
__global__ void k_norm_probs(float* __restrict__ e, const float* __restrict__ g, int N) {
    float inv = 1.0f / g[1];
    for (int i = blockIdx.x * blockDim.x + threadIdx.x; i < N; i += gridDim.x * blockDim.x)
        e[i] *= inv;
}

// ---------------------------------------------------------------------------
// launch
// ---------------------------------------------------------------------------
extern "C" void kernel_launch(void* const* d_in, const int* in_sizes, int n_in,
                              void* d_out, int out_size, void* d_ws, size_t ws_size,
                              hipStream_t stream) {
    const float* x    = (const float*)d_in[0];
    const int*   ei   = (const int*)d_in[1];
    const float* W1   = (const float*)d_in[2];
    const float* b1   = (const float*)d_in[3];
    const float* W2   = (const float*)d_in[4];
    const float* b2   = (const float*)d_in[5];
    const float* W3   = (const float*)d_in[6];
    const float* b3   = (const float*)d_in[7];
    const float* attw = (const float*)d_in[8];

    const int N = in_sizes[0] / 64;       // 100000
    const int E = in_sizes[1] / 2;        // 1000000
    const int* src = ei;
    const int* dst = ei + E;
    const long long NF = (long long)N * 64;

    float* outAtt  = (float*)d_out;       // [N,64] — doubles as feature buffer B
    float* outProb = outAtt + NF;         // [N]    — doubles as scores/exp buffer

    float* ws   = (float*)d_ws;
    float* dinv = ws;                     // N floats
    float* bufA = ws + ((N + 255) & ~255);// N*64 floats (GEMM output)
    float* part = bufA + NF;              // 256 floats (reduction partials)
    float* gred = part + 256;             // 2 floats: [0]=max, [1]=sum

    const int T = 256;
    dim3 blk(T);

    // --- degree / symmetric normalization ---
    k_fill<<<dim3(1024), blk, 0, stream>>>(dinv, 1.0f, (long long)N);   // self loop
    k_deg<<<dim3((E + T - 1) / T), blk, 0, stream>>>(dst, dinv, E);
    k_rsqrt<<<dim3((N + T - 1) / T), blk, 0, stream>>>(dinv, N);

    const int gemmGrid = (N + 127) / 128;                 // 8 waves x 16 rows per block
    const int aggGrid  = (E + N + 7) / 8;                 // 8 edges per block

    // --- layer 1 ---
    k_gemm64<<<dim3(gemmGrid), blk, 0, stream>>>(x, W1, bufA, N);
    k_fill<<<dim3(4096), blk, 0, stream>>>(outAtt, 0.0f, NF);
    k_aggregate<<<dim3(aggGrid), blk, 0, stream>>>(bufA, src, dst, dinv, outAtt, E, N);
    k_bias_relu<<<dim3(4096), blk, 0, stream>>>(outAtt, b1, NF);

    // --- layer 2 ---
    k_gemm64<<<dim3(gemmGrid), blk, 0, stream>>>(outAtt, W2, bufA, N);
    k_fill<<<dim3(4096), blk, 0, stream>>>(outAtt, 0.0f, NF);
    k_aggregate<<<dim3(aggGrid), blk, 0, stream>>>(bufA, src, dst, dinv, outAtt, E, N);
    k_bias_relu<<<dim3(4096), blk, 0, stream>>>(outAtt, b2, NF);

    // --- layer 3 ---
    k_gemm64<<<dim3(gemmGrid), blk, 0, stream>>>(outAtt, W3, bufA, N);
    k_fill<<<dim3(4096), blk, 0, stream>>>(outAtt, 0.0f, NF);
    k_aggregate<<<dim3(aggGrid), blk, 0, stream>>>(bufA, src, dst, dinv, outAtt, E, N);
    k_bias_relu<<<dim3(4096), blk, 0, stream>>>(outAtt, b3, NF);

    // --- attention softmax over all nodes ---
    k_scores<<<dim3((N + 7) / 8), blk, 0, stream>>>(outAtt, attw, outProb, N);
    k_max_part<<<dim3(256), blk, 0, stream>>>(outProb, part, N);
    k_max_final<<<dim3(1), blk, 0, stream>>>(part, gred, 256);
    k_exp_part<<<dim3(256), blk, 0, stream>>>(outProb, gred, part, N);
    k_sum_final<<<dim3(1), blk, 0, stream>>>(part, gred, 256);
    k_attend<<<dim3(4096), blk, 0, stream>>>(outAtt, outProb, gred, NF);
    k_norm_probs<<<dim3((N + T - 1) / T), blk, 0, stream>>>(outProb, gred, N);
}